// LinearRNN_Layer_83124797047475
// MI455X (gfx1250) — compile-verified
//
#include <hip/hip_runtime.h>
#include <math.h>

// ---------------------------------------------------------------------------
// Linear RNN (Euler-Maruyama scan) for MI455X / gfx1250.
//   E_{t+1} = A E_t + sqrt(dt) * n_t,  A = I + dt*(SC - diag(rowsum(SC)))
// Chunked affine-scan: phase1 (parallel zero-state chunk responses, WMMA),
// A^S by repeated squaring (WMMA GEMMs), phase2 (short sequential carries),
// phase3 (parallel history replay per chunk, WMMA), history streamed coalesced.
// ---------------------------------------------------------------------------

typedef float v2f __attribute__((ext_vector_type(2)));
typedef float v8f __attribute__((ext_vector_type(8)));

#define NR      512      // state dimension
#define UP      17       // LDS pitch (conflict-free for scatter + transpose)
#define S_CHUNK 125      // steps per chunk
#define COLS    16       // chunks (columns) per workgroup

__device__ __forceinline__ v8f wmma4(v2f a, v2f b, v8f c) {
  // D = A(16x4,f32) x B(4x16,f32) + C(16x16,f32)
  return __builtin_amdgcn_wmma_f32_16x16x4_f32(
      /*neg_a=*/false, a, /*neg_b=*/false, b,
      /*c_mod=*/(short)0, c, /*reuse_a=*/false, /*reuse_b=*/false);
}

// --------------------------- A = I + dt*lap_sc ------------------------------
__global__ __launch_bounds__(256) void build_A(const float* __restrict__ SC,
                                               float* __restrict__ A, float dt) {
  __shared__ float red[256];
  const int i = blockIdx.x;       // row
  const int tid = threadIdx.x;
  float s = 0.f;
  for (int j = tid; j < NR; j += 256) s += SC[i * NR + j];
  red[tid] = s;
  __syncthreads();
  for (int off = 128; off > 0; off >>= 1) {
    if (tid < off) red[tid] += red[tid + off];
    __syncthreads();
  }
  const float rs = red[0];
  for (int j = tid; j < NR; j += 256) {
    float v = dt * SC[i * NR + j];
    if (j == i) v += 1.0f - dt * rs;     // A[i][i] = 1 + dt*(SC[i][i] - rowsum)
    A[i * NR + j] = v;
  }
}

// ---------------- pre-swizzle A into WMMA A-fragment order ------------------
// A_sw[((rt*128 + kq)*32 + lane)*2 + j] = A[rt*16 + lane%16][kq*4 + 2*(lane/16) + j]
__global__ __launch_bounds__(256) void swizzle_A(const float* __restrict__ A,
                                                 float* __restrict__ A_sw) {
  const int p = blockIdx.x * 256 + threadIdx.x;   // 0 .. 131071
  const int lane = p & 31;
  const int kq = (p >> 5) & 127;
  const int rt = p >> 12;
  const int row = rt * 16 + (lane & 15);
  const int col = kq * 4 + 2 * (lane >> 4);
  A_sw[2 * p]     = A[row * NR + col];
  A_sw[2 * p + 1] = A[row * NR + col + 1];
}

// ------------------- 512x512x512 f32 GEMM (WMMA), C = X*Y -------------------
__global__ __launch_bounds__(256) void gemm512(const float* __restrict__ X,
                                               const float* __restrict__ Y,
                                               float* __restrict__ C) {
  const int tid = threadIdx.x;
  const int lane = tid & 31;
  const int wave = tid >> 5;
  const int w = blockIdx.x * 8 + wave;   // 1024 tiles total, 128 blocks x 8 waves
  const int mt = w >> 5;
  const int nt = w & 31;
  const int n = lane & 15;
  const int hb = lane >> 4;              // k half select
  const int arow = mt * 16 + n;
  v8f acc = {0.f, 0.f, 0.f, 0.f, 0.f, 0.f, 0.f, 0.f};
  for (int kq = 0; kq < 128; ++kq) {
    const int kk = kq * 4 + 2 * hb;
    v2f a = *(const v2f*)(X + arow * NR + kk);
    v2f b;
    b.x = Y[kk * NR + nt * 16 + n];
    b.y = Y[(kk + 1) * NR + nt * 16 + n];
    acc = wmma4(a, b, acc);
  }
  const int m0 = mt * 16 + 8 * hb;
#pragma unroll
  for (int r = 0; r < 8; ++r) C[(m0 + r) * NR + nt * 16 + n] = acc[r];
}

// ----------------------------- PT = P^T -------------------------------------
__global__ __launch_bounds__(256) void transpose512(const float* __restrict__ P,
                                                    float* __restrict__ PT) {
  const int o = blockIdx.x * 256 + threadIdx.x;   // 0 .. 262143
  const int k = o >> 9;
  const int i = o & (NR - 1);
  PT[o] = P[i * NR + k];
}

// ------------- per-chunk recurrence: U <- A*U + sqrt(dt)*noise --------------
// phase==1: U0 = sqrt(dt)*n_{mS}, run j=1..S-1, emit d_m -> dst[(chunk)*512+i]
// phase==3: U0 = Bound[chunk],    run j=0..S-1, emit every step to history
__global__ __launch_bounds__(512) void chunk_kernel(
    const float* __restrict__ A_sw, const float* __restrict__ noise,
    const float* __restrict__ bound, float* __restrict__ dst,
    float* __restrict__ fin, int S, int T, int phase, float sqrt_dt) {
  __shared__ float U[NR * UP];
  const int tid = threadIdx.x;            // 512 threads = 16 waves
  const int lane = tid & 31;
  const int wave = tid >> 5;
  const int cg0 = blockIdx.x * COLS;
  const int n = lane & 15;
  const int hb = lane >> 4;

  // init U (one state row per thread per column)
  if (phase == 1) {
    for (int c = 0; c < COLS; ++c) {
      const long t0 = (long)(cg0 + c) * S;            // uses noise row t0 (j=0)
      U[tid * UP + c] = sqrt_dt * noise[t0 * NR + tid];
    }
  } else {
    for (int c = 0; c < COLS; ++c)
      U[tid * UP + c] = bound[(long)(cg0 + c) * NR + tid];
  }
  __syncthreads();

  const int rt0 = wave * 2;               // 2 row-tiles (32 rows) per wave
  const int rt1 = rt0 + 1;
  const int jstart = (phase == 1) ? 1 : 0;

  for (int j = jstart; j < S; ++j) {
    v8f acc0 = {0.f, 0.f, 0.f, 0.f, 0.f, 0.f, 0.f, 0.f};
    v8f acc1 = {0.f, 0.f, 0.f, 0.f, 0.f, 0.f, 0.f, 0.f};
    for (int kq = 0; kq < 128; ++kq) {
      const int kk = kq * 4 + 2 * hb;
      v2f b;                               // B fragment from LDS (shared by tiles)
      b.x = U[kk * UP + n];
      b.y = U[(kk + 1) * UP + n];
      v2f a0 = *(const v2f*)(A_sw + (((rt0 * 128 + kq) * 32 + lane) << 1));
      v2f a1 = *(const v2f*)(A_sw + (((rt1 * 128 + kq) * 32 + lane) << 1));
      acc0 = wmma4(a0, b, acc0);
      acc1 = wmma4(a1, b, acc1);
    }
    __syncthreads();                       // all waves done reading U
    const int m0 = rt0 * 16 + 8 * hb;
    const int m1 = rt1 * 16 + 8 * hb;
#pragma unroll
    for (int r = 0; r < 8; ++r) {
      U[(m0 + r) * UP + n] = acc0[r];
      U[(m1 + r) * UP + n] = acc1[r];
    }
    __syncthreads();
    // finish: add noise (coalesced) and, for phase 3, stream history rows out
    for (int c = 0; c < COLS; ++c) {
      const long trow = (long)(cg0 + c) * S + j;
      const float u = U[tid * UP + c] + sqrt_dt * noise[trow * NR + tid];
      U[tid * UP + c] = u;
      if (phase == 3) {
        dst[trow * NR + tid] = u;          // state_hist[trow] = E_{trow+1}
        if (trow == (long)T - 1) fin[tid] = u;
      }
    }
    __syncthreads();
  }

  if (phase == 1) {                        // emit d_m (transposed: chunk-major)
    for (int c = 0; c < COLS; ++c)
      dst[(long)(cg0 + c) * NR + tid] = U[tid * UP + c];
  }
}

// ------------- sequential carries: E_{(m+1)S} = A^S E_{mS} + d_m ------------
__global__ __launch_bounds__(512) void carry_kernel(
    const float* __restrict__ PT, const float* __restrict__ D,
    const float* __restrict__ init_state, float* __restrict__ Bound, int M) {
  __shared__ float cv[NR];
  const int tid = threadIdx.x;
  const float c0 = init_state[tid];
  cv[tid] = c0;
  Bound[tid] = c0;                         // Bound[0] = E_0
  __syncthreads();
  for (int m = 0; m < M; ++m) {
    float sum = D[(long)m * NR + tid];
#pragma unroll 8
    for (int k = 0; k < NR; ++k) sum += PT[k * NR + tid] * cv[k];
    __syncthreads();
    cv[tid] = sum;
    Bound[(long)(m + 1) * NR + tid] = sum;
    __syncthreads();
  }
}

// ---------------------------------------------------------------------------
extern "C" void kernel_launch(void* const* d_in, const int* in_sizes, int n_in,
                              void* d_out, int out_size, void* d_ws, size_t ws_size,
                              hipStream_t stream) {
  const float* init_state = (const float*)d_in[0];
  const float* SC         = (const float*)d_in[1];
  const float* noise      = (const float*)d_in[2];
  float* out = (float*)d_out;

  const int T = in_sizes[2] / NR;          // 50000
  const int S = S_CHUNK;                   // 125
  const int M = T / S;                     // 400 chunks (divides exactly)
  const float dt = 1e-4f;
  const float sqrt_dt = sqrtf(dt);

  const size_t NN = (size_t)NR * NR;       // 262144 floats
  float* ws    = (float*)d_ws;
  float* A     = ws;
  float* Asw   = ws + NN;
  float* B1    = ws + 2 * NN;
  float* B2    = ws + 3 * NN;
  float* B3    = ws + 4 * NN;
  float* PT    = ws + 5 * NN;
  float* D     = ws + 6 * NN;              // M*512 floats
  float* Bound = D + (size_t)M * NR;       // (M+1)*512 floats

  build_A<<<NR, 256, 0, stream>>>(SC, A, dt);
  swizzle_A<<<512, 256, 0, stream>>>(A, Asw);

  // A^125 = ((A^5)^5)^5 by repeated squaring/multiplying (9 WMMA GEMMs)
  gemm512<<<128, 256, 0, stream>>>(A,  A,  B1);   // A^2
  gemm512<<<128, 256, 0, stream>>>(B1, B1, B2);   // A^4
  gemm512<<<128, 256, 0, stream>>>(B2, A,  B3);   // A^5
  gemm512<<<128, 256, 0, stream>>>(B3, B3, B1);   // A^10
  gemm512<<<128, 256, 0, stream>>>(B1, B1, B2);   // A^20
  gemm512<<<128, 256, 0, stream>>>(B2, B3, B1);   // A^25
  gemm512<<<128, 256, 0, stream>>>(B1, B1, B2);   // A^50
  gemm512<<<128, 256, 0, stream>>>(B2, B2, B3);   // A^100
  gemm512<<<128, 256, 0, stream>>>(B3, B1, B2);   // A^125
  transpose512<<<1024, 256, 0, stream>>>(B2, PT);

  // phase 1: zero-state chunk responses d_m
  chunk_kernel<<<M / COLS, 512, 0, stream>>>(Asw, noise, nullptr, D, nullptr,
                                             S, T, 1, sqrt_dt);
  // phase 2: carries -> chunk boundary states
  carry_kernel<<<1, 512, 0, stream>>>(PT, D, init_state, Bound, M);
  // phase 3: replay chunks from boundaries, stream full history + final state
  chunk_kernel<<<M / COLS, 512, 0, stream>>>(Asw, noise, Bound, out,
                                             out + (size_t)T * NR, S, T, 3, sqrt_dt);
}